// Net_PILLAR_9096740733110
// MI455X (gfx1250) — compile-verified
//
#include <hip/hip_runtime.h>
#include <math.h>

typedef float v2f __attribute__((ext_vector_type(2)));
typedef float v8f __attribute__((ext_vector_type(8)));

#define RANGE0c (-3.0f)
#define VOXc    (6.0f)
#define EPSBN   (0.001f)
#define EPSMLP  (1e-5f)
#define MAXB    64
#define AST     65   // A staging LDS row stride (pad to kill bank conflicts)
#define WST     65   // W2 LDS row stride

// ---------------------------------------------------------------------------
// fp32 WMMA 16x16x4 wrapper (CDNA5 matrix pipe, numerically exact fp32 path)
// ---------------------------------------------------------------------------
__device__ __forceinline__ v8f wmma_f32_k4(v2f a, v2f b, v8f c) {
  return __builtin_amdgcn_wmma_f32_16x16x4_f32(false, a, false, b, (short)0, c,
                                               false, false);
}

// per-point feature builder (valid-check + 12-dim feature); invalid points are
// fully skipped (they only feed the discarded segment B in the reference)
__device__ __forceinline__ bool point_feat(const float* __restrict__ x,
                                           const int* __restrict__ batch,
                                           const float* __restrict__ pmean,
                                           int i, int* seg, float feat[12]) {
  float p0 = x[6*i+0], p1 = x[6*i+1], p2 = x[6*i+2];
  float e0 = x[6*i+3], e1 = x[6*i+4], e2 = x[6*i+5];
  float cx = floorf((p0 - RANGE0c) / VOXc);
  float cy = floorf((p1 - RANGE0c) / VOXc);
  if (!(cx >= 0.f && cx < 1.f && cy >= 0.f && cy < 1.f)) return false;
  int s = batch[i];
  *seg = s;
  feat[0] = p0; feat[1] = p1; feat[2] = p2;
  feat[3] = p0 - pmean[3*s+0];
  feat[4] = p1 - pmean[3*s+1];
  feat[5] = p2 - pmean[3*s+2];
  feat[6] = p0 - (cx*VOXc + RANGE0c);
  feat[7] = p1 - (cy*VOXc + RANGE0c);
  feat[8] = p2;
  feat[9] = e0; feat[10] = e1; feat[11] = e2;
  return true;
}

// ---------------------------------------------------------------------------
// K1: per-segment count + point-sum (LDS accumulate, ranged flush: batch sorted)
// ---------------------------------------------------------------------------
__global__ void __launch_bounds__(256) k_seg_stats(
    const float* __restrict__ x, const int* __restrict__ batch,
    int N, int per_block, float* __restrict__ cnt, float* __restrict__ psum) {
  __shared__ float scnt[MAXB];
  __shared__ float spsum[MAXB*3];
  int tid = threadIdx.x;
  for (int i = tid; i < MAXB;   i += 256) scnt[i]  = 0.f;
  for (int i = tid; i < MAXB*3; i += 256) spsum[i] = 0.f;
  __syncthreads();
  int start = blockIdx.x * per_block;
  int end   = min(N, start + per_block);
  for (int i = start + tid; i < end; i += 256) {
    float p0 = x[6*i], p1 = x[6*i+1], p2 = x[6*i+2];
    float cx = floorf((p0 - RANGE0c)/VOXc);
    float cy = floorf((p1 - RANGE0c)/VOXc);
    if (cx >= 0.f && cx < 1.f && cy >= 0.f && cy < 1.f) {
      int s = batch[i];
      atomicAdd(&scnt[s], 1.f);
      atomicAdd(&spsum[3*s+0], p0);
      atomicAdd(&spsum[3*s+1], p1);
      atomicAdd(&spsum[3*s+2], p2);
    }
  }
  __syncthreads();
  if (start < end) {
    int smin = batch[start], smax = batch[end-1];
    int tot = (smax - smin + 1) * 4;
    for (int t = tid; t < tot; t += 256) {
      int s = smin + (t >> 2), c = t & 3;
      if (c == 0) { float v = scnt[s];       if (v != 0.f) atomicAdd(&cnt[s], v); }
      else        { float v = spsum[3*s+c-1];if (v != 0.f) atomicAdd(&psum[3*s+c-1], v); }
    }
  }
}

__global__ void k_fin_pmean(const float* __restrict__ cnt,
                            const float* __restrict__ psum, int B,
                            float* __restrict__ pmean, float* __restrict__ nval) {
  int t = threadIdx.x;
  for (int s = t; s <= B; s += blockDim.x) {
    float c = fmaxf(cnt[s], 1.f);
    pmean[3*s+0] = psum[3*s+0] / c;
    pmean[3*s+1] = psum[3*s+1] / c;
    pmean[3*s+2] = psum[3*s+2] / c;
  }
  if (t == 0) {
    float n = 0.f;
    for (int s = 0; s < B; ++s) n += cnt[s];
    nval[0] = fmaxf(n, 1.f);
  }
}

// ---------------------------------------------------------------------------
// K2: h1 = feat@W1, masked sum/sumsq (register accumulate -> LDS -> global)
// ---------------------------------------------------------------------------
__global__ void __launch_bounds__(256) k_h1_stats(
    const float* __restrict__ x, const int* __restrict__ batch,
    const float* __restrict__ W1g, const float* __restrict__ pmean,
    int N, int per_block, float* __restrict__ sum1, float* __restrict__ sumsq1) {
  __shared__ float sW1[12*32];
  __shared__ float sSum[32], sSq[32];
  int tid = threadIdx.x;
  for (int i = tid; i < 384; i += 256) sW1[i] = W1g[i];
  if (tid < 32) { sSum[tid] = 0.f; sSq[tid] = 0.f; }
  __syncthreads();
  int start = blockIdx.x * per_block;
  int end   = min(N, start + per_block);
  float ls[32], lq[32];
  #pragma unroll
  for (int j = 0; j < 32; ++j) { ls[j] = 0.f; lq[j] = 0.f; }
  for (int i = start + tid; i < end; i += 256) {
    int s; float feat[12];
    if (!point_feat(x, batch, pmean, i, &s, feat)) continue;
    #pragma unroll
    for (int j = 0; j < 32; ++j) {
      float h = 0.f;
      #pragma unroll
      for (int k = 0; k < 12; ++k) h += feat[k] * sW1[k*32 + j];
      ls[j] += h; lq[j] += h*h;
    }
  }
  #pragma unroll
  for (int j = 0; j < 32; ++j) {
    atomicAdd(&sSum[j], ls[j]);
    atomicAdd(&sSq[j],  lq[j]);
  }
  __syncthreads();
  if (tid < 32)       atomicAdd(&sum1[tid],      sSum[tid]);
  else if (tid < 64)  atomicAdd(&sumsq1[tid-32], sSq[tid-32]);
}

// BN finalize: scale = gamma*rsqrt(v+eps); shift = beta - m*scale
__global__ void k_fin_bn(const float* __restrict__ sum, const float* __restrict__ sq,
                         const float* __restrict__ gamma, const float* __restrict__ beta,
                         const float* __restrict__ nval, float eps, int C,
                         float* __restrict__ scale, float* __restrict__ shift) {
  int j = threadIdx.x;
  if (j < C) {
    float n = nval[0];
    float m = sum[j] / n;
    float v = sq[j] / n - m*m;
    float sc = gamma[j] * rsqrtf(v + eps);
    scale[j] = sc;
    shift[j] = beta[j] - m*sc;
  }
}

// ---------------------------------------------------------------------------
// K3: segment max of relu(bn1(h1)) -> hmax (uint-bit atomicMax, values >= 0)
// ---------------------------------------------------------------------------
__global__ void __launch_bounds__(256) k_hmax(
    const float* __restrict__ x, const int* __restrict__ batch,
    const float* __restrict__ W1g, const float* __restrict__ pmean,
    const float* __restrict__ sc1g, const float* __restrict__ sh1g,
    int N, int per_block, unsigned* __restrict__ hmax) {
  __shared__ float sW1[12*32];
  __shared__ float sc1[32], sh1[32];
  __shared__ unsigned sMax[MAXB*32];
  int tid = threadIdx.x;
  for (int i = tid; i < 384; i += 256) sW1[i] = W1g[i];
  if (tid < 32) { sc1[tid] = sc1g[tid]; sh1[tid] = sh1g[tid]; }
  for (int i = tid; i < MAXB*32; i += 256) sMax[i] = 0u;
  __syncthreads();
  int start = blockIdx.x * per_block;
  int end   = min(N, start + per_block);
  for (int i = start + tid; i < end; i += 256) {
    int s; float feat[12];
    if (!point_feat(x, batch, pmean, i, &s, feat)) continue;
    #pragma unroll
    for (int j = 0; j < 32; ++j) {
      float h = 0.f;
      #pragma unroll
      for (int k = 0; k < 12; ++k) h += feat[k] * sW1[k*32 + j];
      float y = fmaxf(h*sc1[j] + sh1[j], 0.f);
      atomicMax(&sMax[s*32 + j], __float_as_uint(y));
    }
  }
  __syncthreads();
  if (start < end) {
    int smin = batch[start], smax = batch[end-1];
    int tot = (smax - smin + 1) * 32;
    for (int t = tid; t < tot; t += 256) {
      int s = smin + (t >> 5), j = t & 31;
      unsigned v = sMax[s*32 + j];
      if (v) atomicMax(&hmax[s*32 + j], v);
    }
  }
}

// ---------------------------------------------------------------------------
// K4/K5 shared WMMA core: per wave, stage 32 rows of A=[h1_bn | hmax[seg]]
// into LDS (stride 65), then 2 M-tiles x 4 N-tiles x 16 K-steps of
// V_WMMA_F32_16X16X4_F32 against W2 (64x64) held in LDS.
// K4 accumulates masked column sum/sumsq of h2; K5 applies bn2+relu and does
// per-(segment,col) LDS atomicMax -> global pmax.
// ---------------------------------------------------------------------------
__global__ void __launch_bounds__(256) k_gemm_stats(
    const float* __restrict__ x, const int* __restrict__ batch,
    const float* __restrict__ W1g, const float* __restrict__ pmean,
    const float* __restrict__ sc1g, const float* __restrict__ sh1g,
    const float* __restrict__ W2g, const unsigned* __restrict__ hmaxg,
    int N, int per_block, float* __restrict__ sum2, float* __restrict__ sumsq2) {
  __shared__ float sW1[12*32];
  __shared__ float sc1[32], sh1[32];
  __shared__ float sW2[64*WST];
  __shared__ float sA[8*32*AST];
  __shared__ int   sValid[256];
  __shared__ float sSum[64], sSq[64];
  const int tid = threadIdx.x;
  for (int i = tid; i < 384; i += 256) sW1[i] = W1g[i];
  if (tid < 32) { sc1[tid] = sc1g[tid]; sh1[tid] = sh1g[tid]; }
  for (int i = tid; i < 4096; i += 256) sW2[(i >> 6)*WST + (i & 63)] = W2g[i];
  if (tid < 64) { sSum[tid] = 0.f; sSq[tid] = 0.f; }
  __syncthreads();

  const int start = blockIdx.x * per_block;
  const int end   = min(N, start + per_block);
  const int iters = (per_block + 255) / 256;     // uniform across the block
  const int wave = tid >> 5, lane = tid & 31;
  const int lh = lane >> 4, ll = lane & 15;
  float* Aw = &sA[wave*32*AST];
  float sums[4] = {0.f,0.f,0.f,0.f};
  float sqs[4]  = {0.f,0.f,0.f,0.f};
  const v8f vzero = {0.f,0.f,0.f,0.f,0.f,0.f,0.f,0.f};

  for (int it = 0; it < iters; ++it) {
    int row = start + it*256 + tid;
    bool v = false;
    if (row < end) {
      int s; float feat[12];
      if (point_feat(x, batch, pmean, row, &s, feat)) {
        v = true;
        #pragma unroll
        for (int j = 0; j < 32; ++j) {
          float h = 0.f;
          #pragma unroll
          for (int k = 0; k < 12; ++k) h += feat[k] * sW1[k*32 + j];
          Aw[lane*AST + j] = fmaxf(h*sc1[j] + sh1[j], 0.f);
        }
        #pragma unroll
        for (int j = 0; j < 32; ++j)
          Aw[lane*AST + 32 + j] = __uint_as_float(hmaxg[s*32 + j]);
      }
    }
    if (!v) {
      #pragma unroll
      for (int j = 0; j < 64; ++j) Aw[lane*AST + j] = 0.f;
    }
    sValid[tid] = v ? 1 : 0;
    __syncthreads();

    v8f acc[2][4];
    #pragma unroll
    for (int mt = 0; mt < 2; ++mt)
      #pragma unroll
      for (int nt = 0; nt < 4; ++nt) acc[mt][nt] = vzero;

    #pragma unroll
    for (int k0 = 0; k0 < 64; k0 += 4) {
      int ka = k0 + 2*lh;
      v2f a0, a1;
      a0.x = Aw[ll*AST + ka];       a0.y = Aw[ll*AST + ka + 1];
      a1.x = Aw[(ll+16)*AST + ka];  a1.y = Aw[(ll+16)*AST + ka + 1];
      #pragma unroll
      for (int nt = 0; nt < 4; ++nt) {
        v2f b;
        b.x = sW2[ka*WST + nt*16 + ll];
        b.y = sW2[(ka+1)*WST + nt*16 + ll];
        acc[0][nt] = wmma_f32_k4(a0, b, acc[0][nt]);
        acc[1][nt] = wmma_f32_k4(a1, b, acc[1][nt]);
      }
    }

    // masked column statistics: each lane's C-fragment columns are fixed
    #pragma unroll
    for (int mt = 0; mt < 2; ++mt)
      #pragma unroll
      for (int nt = 0; nt < 4; ++nt)
        #pragma unroll
        for (int r = 0; r < 8; ++r) {
          int rl = mt*16 + r + 8*lh;
          float val = acc[mt][nt][r];
          if (sValid[wave*32 + rl]) { sums[nt] += val; sqs[nt] += val*val; }
        }
    __syncthreads();
  }

  #pragma unroll
  for (int nt = 0; nt < 4; ++nt) {
    int col = nt*16 + ll;
    atomicAdd(&sSum[col], sums[nt]);
    atomicAdd(&sSq[col],  sqs[nt]);
  }
  __syncthreads();
  if (tid < 64) {
    atomicAdd(&sum2[tid],   sSum[tid]);
    atomicAdd(&sumsq2[tid], sSq[tid]);
  }
}

__global__ void __launch_bounds__(256) k_gemm_max(
    const float* __restrict__ x, const int* __restrict__ batch,
    const float* __restrict__ W1g, const float* __restrict__ pmean,
    const float* __restrict__ sc1g, const float* __restrict__ sh1g,
    const float* __restrict__ W2g, const unsigned* __restrict__ hmaxg,
    const float* __restrict__ sc2g, const float* __restrict__ sh2g,
    int N, int per_block, unsigned* __restrict__ pmax) {
  __shared__ float sW1[12*32];
  __shared__ float sc1[32], sh1[32];
  __shared__ float sc2[64], sh2[64];
  __shared__ float sW2[64*WST];
  __shared__ float sA[8*32*AST];
  __shared__ int   sValid[256];
  __shared__ int   sSeg[256];
  __shared__ unsigned sPmax[MAXB*64];
  const int tid = threadIdx.x;
  for (int i = tid; i < 384; i += 256) sW1[i] = W1g[i];
  if (tid < 32) { sc1[tid] = sc1g[tid]; sh1[tid] = sh1g[tid]; }
  if (tid < 64) { sc2[tid] = sc2g[tid]; sh2[tid] = sh2g[tid]; }
  for (int i = tid; i < 4096; i += 256) sW2[(i >> 6)*WST + (i & 63)] = W2g[i];
  for (int i = tid; i < MAXB*64; i += 256) sPmax[i] = 0u;
  __syncthreads();

  const int start = blockIdx.x * per_block;
  const int end   = min(N, start + per_block);
  const int iters = (per_block + 255) / 256;
  const int wave = tid >> 5, lane = tid & 31;
  const int lh = lane >> 4, ll = lane & 15;
  float* Aw = &sA[wave*32*AST];
  const v8f vzero = {0.f,0.f,0.f,0.f,0.f,0.f,0.f,0.f};

  for (int it = 0; it < iters; ++it) {
    int row = start + it*256 + tid;
    bool v = false; int seg = 0;
    if (row < end) {
      int s; float feat[12];
      if (point_feat(x, batch, pmean, row, &s, feat)) {
        v = true; seg = s;
        #pragma unroll
        for (int j = 0; j < 32; ++j) {
          float h = 0.f;
          #pragma unroll
          for (int k = 0; k < 12; ++k) h += feat[k] * sW1[k*32 + j];
          Aw[lane*AST + j] = fmaxf(h*sc1[j] + sh1[j], 0.f);
        }
        #pragma unroll
        for (int j = 0; j < 32; ++j)
          Aw[lane*AST + 32 + j] = __uint_as_float(hmaxg[s*32 + j]);
      }
    }
    if (!v) {
      #pragma unroll
      for (int j = 0; j < 64; ++j) Aw[lane*AST + j] = 0.f;
    }
    sValid[tid] = v ? 1 : 0;
    sSeg[tid] = seg;
    __syncthreads();

    v8f acc[2][4];
    #pragma unroll
    for (int mt = 0; mt < 2; ++mt)
      #pragma unroll
      for (int nt = 0; nt < 4; ++nt) acc[mt][nt] = vzero;

    #pragma unroll
    for (int k0 = 0; k0 < 64; k0 += 4) {
      int ka = k0 + 2*lh;
      v2f a0, a1;
      a0.x = Aw[ll*AST + ka];       a0.y = Aw[ll*AST + ka + 1];
      a1.x = Aw[(ll+16)*AST + ka];  a1.y = Aw[(ll+16)*AST + ka + 1];
      #pragma unroll
      for (int nt = 0; nt < 4; ++nt) {
        v2f b;
        b.x = sW2[ka*WST + nt*16 + ll];
        b.y = sW2[(ka+1)*WST + nt*16 + ll];
        acc[0][nt] = wmma_f32_k4(a0, b, acc[0][nt]);
        acc[1][nt] = wmma_f32_k4(a1, b, acc[1][nt]);
      }
    }

    #pragma unroll
    for (int mt = 0; mt < 2; ++mt)
      #pragma unroll
      for (int nt = 0; nt < 4; ++nt) {
        int col = nt*16 + ll;
        #pragma unroll
        for (int r = 0; r < 8; ++r) {
          int rl = mt*16 + r + 8*lh;
          if (sValid[wave*32 + rl]) {
            float y = fmaxf(acc[mt][nt][r]*sc2[col] + sh2[col], 0.f);
            atomicMax(&sPmax[sSeg[wave*32 + rl]*64 + col], __float_as_uint(y));
          }
        }
      }
    __syncthreads();
  }

  if (start < end) {
    int smin = batch[start], smax = batch[end-1];
    int tot = (smax - smin + 1) * 64;
    for (int t = tid; t < tot; t += 256) {
      int s = smin + (t >> 6), c = t & 63;
      unsigned vb = sPmax[s*64 + c];
      if (vb) atomicMax(&pmax[s*64 + c], vb);
    }
  }
}

// ---------------------------------------------------------------------------
// Head: z = relu(bn(p @ Wc^T)); d = z2 - z1   (4 channels x 64 rows per block)
// ---------------------------------------------------------------------------
__global__ void __launch_bounds__(256) k_head1(
    const float* __restrict__ p1, const float* __restrict__ p2,
    const float* __restrict__ Wc, const float* __restrict__ gc,
    const float* __restrict__ bc, int B, int CH, float* __restrict__ dbuf) {
  __shared__ float sP1[64*64], sP2[64*64];
  __shared__ float red[256], red2[256];
  int tid = threadIdx.x;
  for (int i = tid; i < B*64; i += 256) { sP1[i] = p1[i]; sP2[i] = p2[i]; }
  __syncthreads();
  int row = tid & 63, cl = tid >> 6;
  int c = blockIdx.x*4 + cl;
  bool act = (row < B) && (c < CH);
  float q[2] = {0.f, 0.f};
  if (act) {
    for (int k = 0; k < 64; ++k) {
      float w = Wc[c*64 + k];
      q[0] += sP1[row*64 + k] * w;
      q[1] += sP2[row*64 + k] * w;
    }
  }
  float gcv = (c < CH) ? gc[c] : 0.f;
  float bcv = (c < CH) ? bc[c] : 0.f;
  float z[2];
  for (int br = 0; br < 2; ++br) {
    red[tid]  = act ? q[br]       : 0.f;
    red2[tid] = act ? q[br]*q[br] : 0.f;
    __syncthreads();
    for (int off = 32; off > 0; off >>= 1) {
      if (row < off) { red[tid] += red[tid+off]; red2[tid] += red2[tid+off]; }
      __syncthreads();
    }
    float m  = red[cl*64] / (float)B;
    float vv = red2[cl*64] / (float)B - m*m;
    z[br] = fmaxf((q[br] - m)*rsqrtf(vv + EPSBN)*gcv + bcv, 0.f);
    __syncthreads();
  }
  if (act) dbuf[row*CH + c] = z[1] - z[0];
}

// h = bn(relu(d @ Wm1 + bm1), EPS_MLP)
__global__ void __launch_bounds__(1024) k_head2(
    const float* __restrict__ dbuf, const float* __restrict__ Wm1,
    const float* __restrict__ bm1, const float* __restrict__ gm,
    const float* __restrict__ bm, int B, int CH, float* __restrict__ hbuf) {
  __shared__ float sR[64*64];
  __shared__ float sSc[64], sSh[64];
  int tid = threadIdx.x;
  for (int o = tid; o < 64*64; o += 1024) {
    int i = o >> 6, j = o & 63;
    float a = 0.f;
    if (i < B) {
      a = bm1[j];
      for (int c = 0; c < CH; ++c) a += dbuf[i*CH + c] * Wm1[c*64 + j];
      a = fmaxf(a, 0.f);
    }
    sR[o] = a;
  }
  __syncthreads();
  if (tid < 64) {
    float s = 0.f, qq = 0.f;
    for (int i = 0; i < B; ++i) { float r = sR[i*64 + tid]; s += r; qq += r*r; }
    float m = s/(float)B, vv = qq/(float)B - m*m;
    float sc = gm[tid]*rsqrtf(vv + EPSMLP);
    sSc[tid] = sc; sSh[tid] = bm[tid] - m*sc;
  }
  __syncthreads();
  for (int o = tid; o < B*64; o += 1024) {
    int j = o & 63;
    hbuf[o] = sR[o]*sSc[j] + sSh[j];
  }
}

// logits + log_softmax
__global__ void k_head3(const float* __restrict__ hbuf, const float* __restrict__ Wm2,
                        const float* __restrict__ bm2, int B, float* __restrict__ out) {
  int i = blockIdx.x*blockDim.x + threadIdx.x;
  if (i >= B) return;
  float l[5];
  #pragma unroll
  for (int j = 0; j < 5; ++j) {
    float a = bm2[j];
    for (int k = 0; k < 64; ++k) a += hbuf[i*64 + k] * Wm2[k*5 + j];
    l[j] = a;
  }
  float m = l[0];
  #pragma unroll
  for (int j = 1; j < 5; ++j) m = fmaxf(m, l[j]);
  float s = 0.f;
  #pragma unroll
  for (int j = 0; j < 5; ++j) s += expf(l[j] - m);
  float lse = logf(s);
  #pragma unroll
  for (int j = 0; j < 5; ++j) out[i*5 + j] = l[j] - m - lse;
}

// ---------------------------------------------------------------------------
extern "C" void kernel_launch(void* const* d_in, const int* in_sizes, int n_in,
                              void* d_out, int out_size, void* d_ws, size_t ws_size,
                              hipStream_t stream) {
  (void)n_in; (void)out_size; (void)ws_size;
  const float* x    = (const float*)d_in[0];
  const float* x2   = (const float*)d_in[1];
  const int*   bt1  = (const int*)d_in[2];
  const int*   bt2  = (const int*)d_in[3];
  const float* W1   = (const float*)d_in[5];
  const float* g1   = (const float*)d_in[6];
  const float* bb1  = (const float*)d_in[7];
  const float* W2   = (const float*)d_in[8];
  const float* g2   = (const float*)d_in[9];
  const float* bb2  = (const float*)d_in[10];
  const float* Wc   = (const float*)d_in[11];
  const float* gc   = (const float*)d_in[12];
  const float* bc   = (const float*)d_in[13];
  const float* Wm1  = (const float*)d_in[14];
  const float* bm1  = (const float*)d_in[15];
  const float* gm   = (const float*)d_in[16];
  const float* bm   = (const float*)d_in[17];
  const float* Wm2  = (const float*)d_in[18];
  const float* bm2  = (const float*)d_in[19];
  const int N  = in_sizes[0] / 6;
  const int B  = in_sizes[4];
  const int CH = in_sizes[11] / 64;     // 1024

  float* wsf = (float*)d_ws;
  // workspace layout (floats)
  size_t oCnt   = 0;
  size_t oPsum  = oCnt   + (size_t)(B+1);
  size_t oPmean = oPsum  + 3*(size_t)(B+1);
  size_t oNval  = oPmean + 3*(size_t)(B+1);
  size_t oSum1  = oNval  + 1;
  size_t oSq1   = oSum1  + 32;
  size_t oSc1   = oSq1   + 32;
  size_t oSh1   = oSc1   + 32;
  size_t oSum2  = oSh1   + 32;
  size_t oSq2   = oSum2  + 64;
  size_t oSc2   = oSq2   + 64;
  size_t oSh2   = oSc2   + 64;
  size_t oHmax  = oSh2   + 64;
  size_t oStatsEnd = oHmax + 32*(size_t)(B+1);
  size_t oPmax1 = oStatsEnd;
  size_t oPmax2 = oPmax1 + 64*(size_t)B;
  size_t oDbuf  = oPmax2 + 64*(size_t)B;
  size_t oHbuf  = oDbuf  + (size_t)B * CH;

  const int NB1 = 512, NB2 = 1024;
  const int pb1 = (N + NB1 - 1) / NB1;
  const int pb2 = (N + NB2 - 1) / NB2;

  hipMemsetAsync((void*)(wsf + oPmax1), 0, 2*64*(size_t)B*sizeof(float), stream);

  auto run_branch = [&](const float* xb, const int* bb, unsigned* pmax) {
    hipMemsetAsync((void*)(wsf + oCnt), 0, (oStatsEnd - oCnt)*sizeof(float), stream);
    k_seg_stats<<<NB1, 256, 0, stream>>>(xb, bb, N, pb1, wsf+oCnt, wsf+oPsum);
    k_fin_pmean<<<1, 256, 0, stream>>>(wsf+oCnt, wsf+oPsum, B, wsf+oPmean, wsf+oNval);
    k_h1_stats<<<NB1, 256, 0, stream>>>(xb, bb, W1, wsf+oPmean, N, pb1,
                                        wsf+oSum1, wsf+oSq1);
    k_fin_bn<<<1, 32, 0, stream>>>(wsf+oSum1, wsf+oSq1, g1, bb1, wsf+oNval,
                                   EPSBN, 32, wsf+oSc1, wsf+oSh1);
    k_hmax<<<NB1, 256, 0, stream>>>(xb, bb, W1, wsf+oPmean, wsf+oSc1, wsf+oSh1,
                                    N, pb1, (unsigned*)(wsf+oHmax));
    k_gemm_stats<<<NB2, 256, 0, stream>>>(xb, bb, W1, wsf+oPmean, wsf+oSc1,
                                          wsf+oSh1, W2,
                                          (const unsigned*)(wsf+oHmax),
                                          N, pb2, wsf+oSum2, wsf+oSq2);
    k_fin_bn<<<1, 64, 0, stream>>>(wsf+oSum2, wsf+oSq2, g2, bb2, wsf+oNval,
                                   EPSBN, 64, wsf+oSc2, wsf+oSh2);
    k_gemm_max<<<NB2, 256, 0, stream>>>(xb, bb, W1, wsf+oPmean, wsf+oSc1,
                                        wsf+oSh1, W2,
                                        (const unsigned*)(wsf+oHmax),
                                        wsf+oSc2, wsf+oSh2, N, pb2, pmax);
  };

  run_branch(x,  bt1, (unsigned*)(wsf + oPmax1));
  run_branch(x2, bt2, (unsigned*)(wsf + oPmax2));

  // head (pmax bits are non-negative float bit patterns -> read directly as float)
  k_head1<<<(CH + 3)/4, 256, 0, stream>>>(wsf+oPmax1, wsf+oPmax2, Wc, gc, bc,
                                          B, CH, wsf+oDbuf);
  k_head2<<<1, 1024, 0, stream>>>(wsf+oDbuf, Wm1, bm1, gm, bm, B, CH, wsf+oHbuf);
  k_head3<<<(B + 63)/64, 64, 0, stream>>>(wsf+oHbuf, Wm2, bm2, B, (float*)d_out);
}